// VideoPanopticPredictionStitcher_54752243089795
// MI455X (gfx1250) — compile-verified
//
#include <hip/hip_runtime.h>
#include <stdint.h>

// Video panoptic stitcher for MI455X (gfx1250, wave32).
//   K1 init workspace
//   K2 double-buffered async-LDS staged area histograms + (c,n) pair hash counts
//   K3 IoU scoring + packed atomicMax argmax per concat-id and per-next-id
//   K4 mutual best-match -> mapping[20000]
//   K5 output gather, double-buffered global_load_async_to_lds_b128 staging
// CDNA5-specific paths: async global->LDS data movers (ASYNCcnt),
// s_wait_asynccnt pipelining, global_prefetch_b8.

#define LABEL_DIVISOR 1000
#define OVERLAP_OFFSET 500
#define MAX_ID 20000          // (NUM_CATEGORIES + 1) * LABEL_DIVISOR
#define HSIZE 65536
#define HMASK (HSIZE - 1)
#define HEMPTY 0xFFFFFFFFu

__device__ __forceinline__ int transform_next(int np) {
  int cat  = np / LABEL_DIVISOR;
  int inst = np - cat * LABEL_DIVISOR;
  return cat * LABEL_DIVISOR + (inst > 0 ? inst + OVERLAP_OFFSET : inst);
}

__device__ __forceinline__ int clamp_id(int v) {
  v = v < 0 ? 0 : v;
  return v >= MAX_ID ? (MAX_ID - 1) : v;
}

// generic LDS pointer low 32 bits == LDS byte offset (ISA aperture rule)
__device__ __forceinline__ uint32_t lds_off(const void* p) {
  return (uint32_t)(uintptr_t)p;
}

__device__ __forceinline__ void async_b32(uint32_t lds, const void* g) {
  uint64_t ga = (uint64_t)(uintptr_t)g;
  asm volatile("global_load_async_to_lds_b32 %0, %1, off"
               :: "v"(lds), "v"(ga) : "memory");
}

__device__ __forceinline__ void async_b128(uint32_t lds, const void* g) {
  uint64_t ga = (uint64_t)(uintptr_t)g;
  asm volatile("global_load_async_to_lds_b128 %0, %1, off"
               :: "v"(lds), "v"(ga) : "memory");
}

// Length of the run of consecutive in-bounds lanes starting at a leader lane.
__device__ __forceinline__ int run_len(bool leader, unsigned lane, unsigned act) {
  unsigned b    = (unsigned)__ballot(leader);
  unsigned stop = b | ~act;                      // next leader or first OOB lane
  unsigned rest = stop & (0xFFFFFFFEu << lane);  // bits strictly above `lane`
  int next = rest ? (int)__builtin_ctz(rest) : 32;
  return next - (int)lane;
}

__device__ __forceinline__ void hash_add(uint32_t* hkeys, uint32_t* hcnt,
                                         uint32_t key, uint32_t cnt) {
  uint32_t slot = ((key * 2654435761u) >> 14) & HMASK;
  for (int probe = 0; probe < HSIZE; ++probe) {
    uint32_t old = atomicCAS(&hkeys[slot], HEMPTY, key);
    if (old == HEMPTY || old == key) { atomicAdd(&hcnt[slot], cnt); return; }
    slot = (slot + 1) & HMASK;
  }
}

// ---------------- K1: init workspace ----------------
__global__ void init_ws_kernel(unsigned long long* best_c, unsigned long long* best_n,
                               uint32_t* area_c, uint32_t* area_n, int* mapping,
                               uint32_t* hkeys, uint32_t* hcnt) {
  int i = blockIdx.x * blockDim.x + threadIdx.x;
  if (i < MAX_ID) {
    best_c[i] = 0ull; best_n[i] = 0ull;
    area_c[i] = 0u;   area_n[i] = 0u;
    mapping[i] = i;
  }
  if (i < HSIZE) { hkeys[i] = HEMPTY; hcnt[i] = 0u; }
}

// ---------------- K2: areas + pair intersections ----------------
// Double-buffered async staging: each wave issues 2 async b32 loads per chunk;
// s_wait_asynccnt 0x2 after issuing chunk k+1 guarantees chunk k is resident
// (async loads retire in order).
__global__ __launch_bounds__(256)
void hist_kernel(const int* __restrict__ concat, const int* __restrict__ next,
                 uint32_t* area_c, uint32_t* area_n,
                 uint32_t* hkeys, uint32_t* hcnt, int N) {
  __shared__ int tc[2][256];
  __shared__ int tn[2][256];
  const int tid = threadIdx.x;
  const unsigned lane = (unsigned)tid & 31u;
  uint32_t lc[2] = { lds_off(&tc[0][tid]), lds_off(&tc[1][tid]) };
  uint32_t ln[2] = { lds_off(&tn[0][tid]), lds_off(&tn[1][tid]) };

  const long long stride = (long long)gridDim.x * 256;
  const long long base0  = (long long)blockIdx.x * 256;
  int buf = 0;

  if (base0 < (long long)N) {  // uniform per block
    long long a = base0 + tid; if (a > (long long)N - 1) a = (long long)N - 1;
    async_b32(lc[0], concat + a);
    async_b32(ln[0], next + a);
  }
  for (long long base = base0; base < (long long)N; base += stride) {
    long long nb = base + stride;
    if (nb < (long long)N) {   // uniform
      long long a = nb + tid; if (a > (long long)N - 1) a = (long long)N - 1;
      // prefetch the same next-chunk concat address that is async-loaded below
      __builtin_prefetch(concat + a, 0, 0);
      async_b32(lc[buf ^ 1], concat + a);
      async_b32(ln[buf ^ 1], next + a);
      asm volatile("s_wait_asynccnt 0x2" ::: "memory");
    } else {
      asm volatile("s_wait_asynccnt 0x0" ::: "memory");
    }

    long long i = base + tid;
    bool inb = i < (long long)N;               // OOB lanes are a wave suffix
    int c = clamp_id(tc[buf][tid]);
    int n = clamp_id(transform_next(tn[buf][tid]));
    unsigned act = (unsigned)__ballot(inb);

    // wave-aggregated area_c
    int pc = __shfl_up(c, 1);
    bool lead_c = (lane == 0u) || (pc != c);
    int cnt_c = run_len(lead_c, lane, act);
    if (lead_c && inb) atomicAdd(&area_c[c], (uint32_t)cnt_c);

    // wave-aggregated area_n
    int pn = __shfl_up(n, 1);
    bool lead_n = (lane == 0u) || (pn != n);
    int cnt_n = run_len(lead_n, lane, act);
    if (lead_n && inb) atomicAdd(&area_n[n], (uint32_t)cnt_n);

    // valid pair: same category, both instance ids nonzero
    bool valid = inb &&
                 (c / LABEL_DIVISOR == n / LABEL_DIVISOR) &&
                 (c % LABEL_DIVISOR != 0) && (n % LABEL_DIVISOR != 0);
    uint32_t key = valid ? (uint32_t)(c * MAX_ID + n) : HEMPTY;
    uint32_t pk = (uint32_t)__shfl_up((int)key, 1);
    bool lead_p = (lane == 0u) || (pk != key);
    int cnt_p = run_len(lead_p, lane, act);
    if (lead_p && key != HEMPTY) hash_add(hkeys, hcnt, key, (uint32_t)cnt_p);

    buf ^= 1;
  }
}

// ---------------- K3: IoU + per-id argmax ----------------
__global__ void score_kernel(const uint32_t* __restrict__ hkeys,
                             const uint32_t* __restrict__ hcnt,
                             const uint32_t* __restrict__ area_c,
                             const uint32_t* __restrict__ area_n,
                             unsigned long long* best_c, unsigned long long* best_n) {
  int s = blockIdx.x * blockDim.x + threadIdx.x;
  if (s >= HSIZE) return;
  uint32_t key = hkeys[s];
  if (key == HEMPTY) return;
  uint32_t inter = hcnt[s];
  uint32_t c = key / MAX_ID;
  uint32_t n = key % MAX_ID;
  uint32_t uni = area_c[c] + area_n[n] - inter;
  float iou = (float)inter / (float)(uni ? uni : 1u);
  // positive-float bit pattern is order preserving; tie-break on larger partner id
  unsigned long long hi = ((unsigned long long)__float_as_uint(iou)) << 32;
  atomicMax(&best_c[c], hi | (unsigned long long)n);
  atomicMax(&best_n[n], hi | (unsigned long long)c);
}

// ---------------- K4: mutual best-match mapping ----------------
__global__ void map_kernel(const unsigned long long* __restrict__ best_c,
                           const unsigned long long* __restrict__ best_n,
                           int* __restrict__ mapping) {
  int n = blockIdx.x * blockDim.x + threadIdx.x;
  if (n >= MAX_ID) return;
  int m = n;
  unsigned long long bn = best_n[n];
  if (bn != 0ull) {
    int cstar = (int)(bn & 0xFFFFFFFFull);
    if (cstar >= 0 && cstar < MAX_ID) {
      unsigned long long bc = best_c[cstar];
      if (bc != 0ull && (int)(bc & 0xFFFFFFFFull) == n) m = cstar;
    }
  }
  mapping[n] = m;
}

// ---------------- K5: async-staged gather + output ----------------
#define TILE_INTS 1024  // 256 lanes x 16B async b128 loads = 4KB per chunk
__global__ __launch_bounds__(256)
void out_kernel(const int* __restrict__ next, const int* __restrict__ mapping,
                float* __restrict__ out, int N) {
  __shared__ int tile[2][TILE_INTS];
  const int tid = threadIdx.x;
  uint32_t lds[2] = { lds_off(&tile[0][tid * 4]), lds_off(&tile[1][tid * 4]) };

  const long long cstride = (long long)gridDim.x * TILE_INTS;
  const long long base0   = (long long)blockIdx.x * TILE_INTS;
  int buf = 0;

  if (base0 < (long long)N && N >= 4) {  // uniform per block
    long long a = base0 + (long long)tid * 4;
    if (a > (long long)N - 4) a = (long long)N - 4;
    if (a < 0) a = 0;
    async_b128(lds[0], next + a);
  }
  for (long long base = base0; base < (long long)N; base += cstride) {
    long long nb = base + cstride;
    if (nb < (long long)N && N >= 4) {   // uniform
      long long a = nb + (long long)tid * 4;
      if (a > (long long)N - 4) a = (long long)N - 4;
      if (a < 0) a = 0;
      async_b128(lds[buf ^ 1], next + a);
      asm volatile("s_wait_asynccnt 0x1" ::: "memory");
    } else {
      asm volatile("s_wait_asynccnt 0x0" ::: "memory");
    }

    long long idx = base + (long long)tid * 4;
    bool full = (idx + 3) < (long long)N && N >= 4;
#pragma unroll
    for (int j = 0; j < 4; ++j) {
      long long i = idx + j;
      if (i < (long long)N) {
        int v  = full ? tile[buf][tid * 4 + j] : next[i];
        out[i] = (float)mapping[clamp_id(transform_next(v))];
      }
    }
    buf ^= 1;
  }
}

extern "C" void kernel_launch(void* const* d_in, const int* in_sizes, int n_in,
                              void* d_out, int out_size, void* d_ws, size_t ws_size,
                              hipStream_t stream) {
  (void)n_in; (void)out_size; (void)ws_size;
  const int* concat = (const int*)d_in[0];
  const int* next   = (const int*)d_in[1];
  float* out = (float*)d_out;
  const int N = in_sizes[0];

  // workspace layout (8-byte aligned items first); total ~1.05 MB
  char* ws = (char*)d_ws;
  unsigned long long* best_c = (unsigned long long*)ws; ws += (size_t)MAX_ID * 8;
  unsigned long long* best_n = (unsigned long long*)ws; ws += (size_t)MAX_ID * 8;
  uint32_t* area_c  = (uint32_t*)ws; ws += (size_t)MAX_ID * 4;
  uint32_t* area_n  = (uint32_t*)ws; ws += (size_t)MAX_ID * 4;
  int*      mapping = (int*)ws;      ws += (size_t)MAX_ID * 4;
  uint32_t* hkeys   = (uint32_t*)ws; ws += (size_t)HSIZE * 4;
  uint32_t* hcnt    = (uint32_t*)ws;

  init_ws_kernel<<<(HSIZE + 255) / 256, 256, 0, stream>>>(
      best_c, best_n, area_c, area_n, mapping, hkeys, hcnt);

  hist_kernel<<<1024, 256, 0, stream>>>(
      concat, next, area_c, area_n, hkeys, hcnt, N);

  score_kernel<<<HSIZE / 256, 256, 0, stream>>>(
      hkeys, hcnt, area_c, area_n, best_c, best_n);

  map_kernel<<<(MAX_ID + 255) / 256, 256, 0, stream>>>(best_c, best_n, mapping);

  out_kernel<<<512, 256, 0, stream>>>(next, mapping, out, N);
}